// EM_fusion_51127290692324
// MI455X (gfx1250) — compile-verified
//
#include <hip/hip_runtime.h>

// EM_fusion on MI455X (gfx1250).
//
// softmax over two identical logits is exactly [0.5, 0.5] in fp32, so
//   fused = 0.5*(feature_1 + feature_2)   (bit-identical: *0.5 is exact).
// Pure stream: 3 * 131072*768*4B = 1.21 GB -> ~52 us floor at 23.3 TB/s.
// Memory-bound ~100x over compute; WMMA/TDM inapplicable. Optimal path:
// b128 NT loads/stores, wave32 blocks, branch-free full-tile fast path so
// each wave keeps 8 b128 loads in flight before the first wait.

typedef __attribute__((ext_vector_type(4))) float f32x4;

#define TPB    256          // threads per block (8 wave32 waves)
#define UNROLL 4            // float4s per thread
#define TILE   (TPB * UNROLL)

__global__ __launch_bounds__(TPB) void em_fusion_avg_kernel(
    const f32x4* __restrict__ a4,
    const f32x4* __restrict__ b4,
    f32x4* __restrict__ o4,
    const float* __restrict__ a,
    const float* __restrict__ b,
    float* __restrict__ o,
    size_t n4,   // number of float4 elements
    size_t n)    // total float elements
{
    size_t tile_base = (size_t)blockIdx.x * TILE;
    size_t base      = tile_base + threadIdx.x;

    if (tile_base + TILE <= n4) {
        // -------- Fast path: whole tile in range, no guards --------
        // Issue all 8 NT b128 loads before any use -> max per-wave MLP.
        f32x4 av[UNROLL], bv[UNROLL];
#pragma unroll
        for (int u = 0; u < UNROLL; ++u)
            av[u] = __builtin_nontemporal_load(a4 + base + (size_t)u * TPB);
#pragma unroll
        for (int u = 0; u < UNROLL; ++u)
            bv[u] = __builtin_nontemporal_load(b4 + base + (size_t)u * TPB);
#pragma unroll
        for (int u = 0; u < UNROLL; ++u) {
            f32x4 r = (av[u] + bv[u]) * 0.5f;
            __builtin_nontemporal_store(r, o4 + base + (size_t)u * TPB);
        }
    } else {
        // -------- Slow path: final partial tile, per-access guard --------
#pragma unroll
        for (int u = 0; u < UNROLL; ++u) {
            size_t i = base + (size_t)u * TPB;
            if (i < n4) {
                f32x4 av = __builtin_nontemporal_load(a4 + i);
                f32x4 bv = __builtin_nontemporal_load(b4 + i);
                f32x4 r  = (av + bv) * 0.5f;
                __builtin_nontemporal_store(r, o4 + i);
            }
        }
        // Scalar tail for n % 4 (zero for B=131072, D=768, kept general).
        size_t tail = n - n4 * 4;
        if (blockIdx.x == gridDim.x - 1 && (size_t)threadIdx.x < tail) {
            size_t i = n4 * 4 + threadIdx.x;
            o[i] = (a[i] + b[i]) * 0.5f;
        }
    }
}

extern "C" void kernel_launch(void* const* d_in, const int* in_sizes, int n_in,
                              void* d_out, int out_size, void* d_ws, size_t ws_size,
                              hipStream_t stream) {
    (void)in_sizes; (void)n_in; (void)d_ws; (void)ws_size;

    const float* a = (const float*)d_in[0];   // feature_1, fp32 [B, D]
    const float* b = (const float*)d_in[1];   // feature_2, fp32 [B, D]
    float* o       = (float*)d_out;           // fused,     fp32 [B, D]

    size_t n  = (size_t)out_size;             // 131072 * 768 = 100,663,296
    size_t n4 = n / 4;                        // 25,165,824 float4s

    size_t grid = (n4 + TILE - 1) / TILE;     // 24,576 blocks (exact fit)
    if (grid == 0) grid = 1;

    em_fusion_avg_kernel<<<(uint32_t)grid, TPB, 0, stream>>>(
        (const f32x4*)a, (const f32x4*)b, (f32x4*)o, a, b, o, n4, n);
}